// GNN_17205638988431
// MI455X (gfx1250) — compile-verified
//
#include <hip/hip_runtime.h>
#include <stdint.h>

// ---------------------------------------------------------------------------
// GraphSAGE x3 on gfx1250 (MI455X).
// Phase 0 (once per launch): build reverse-CSR (dst -> list of src).
// Per layer: atomic-free mean aggregation (wave per node), then fused
// dual-GEMM  out = mean @ Wl^T + h @ Wr^T + b  using V_WMMA_F32_16X16X4_F32.
// GEMM: one wave computes a full 16x128 output row-strip (8 accumulators),
// so each A fragment (mean/h) is loaded once and reused for 8 WMMAs.
// ---------------------------------------------------------------------------

#define HIDDEN 128
#define SCAN_THREADS 1024

typedef float v2f __attribute__((ext_vector_type(2)));
typedef float v8f __attribute__((ext_vector_type(8)));

// ---------------- CSR build ----------------

__global__ __launch_bounds__(256) void count_kernel(
    const long long* __restrict__ ei, int* __restrict__ cnt, int E)
{
    int e = blockIdx.x * blockDim.x + threadIdx.x;
    if (e >= E) return;
    int d = (int)ei[(size_t)E + e];       // row 1 = dst
    atomicAdd(&cnt[d], 1);
}

__global__ __launch_bounds__(SCAN_THREADS) void scan_kernel(
    const int* __restrict__ cnt, int* __restrict__ offs, int n)
{
    __shared__ int sm[SCAN_THREADS];
    __shared__ int carry;
    int tid = threadIdx.x;
    if (tid == 0) carry = 0;
    __syncthreads();
    for (int base = 0; base < n; base += SCAN_THREADS) {
        int i = base + tid;
        int v = (i < n) ? cnt[i] : 0;
        sm[tid] = v;
        __syncthreads();
        // Hillis-Steele inclusive scan
        for (int off = 1; off < SCAN_THREADS; off <<= 1) {
            int t = (tid >= off) ? sm[tid - off] : 0;
            __syncthreads();
            sm[tid] += t;
            __syncthreads();
        }
        if (i < n) offs[i] = carry + sm[tid] - v;   // exclusive
        __syncthreads();
        if (tid == SCAN_THREADS - 1) carry += sm[SCAN_THREADS - 1];
        __syncthreads();
    }
    if (tid == 0) offs[n] = carry;
}

__global__ __launch_bounds__(256) void fill_kernel(
    const long long* __restrict__ ei, const int* __restrict__ offs,
    int* __restrict__ cursor, int* __restrict__ nbr, int E)
{
    int e = blockIdx.x * blockDim.x + threadIdx.x;
    if (e >= E) return;
    int s = (int)ei[e];                   // row 0 = src
    int d = (int)ei[(size_t)E + e];       // row 1 = dst
    int p = atomicAdd(&cursor[d], 1);
    nbr[offs[d] + p] = s;
}

// ---------------- per-layer aggregation (atomic-free) ----------------
// one wave32 per node; lane covers 4 consecutive features (32*4 = 128)

__global__ __launch_bounds__(256) void sage_aggregate(
    const float* __restrict__ x, const int* __restrict__ offs,
    const int* __restrict__ nbr, float* __restrict__ mean, int N)
{
    int wave = (blockIdx.x * blockDim.x + threadIdx.x) >> 5;
    int lane = threadIdx.x & 31;
    if (wave >= N) return;
    int beg = offs[wave], end = offs[wave + 1];
    const float4* xv = (const float4*)x;
    float4 acc = make_float4(0.f, 0.f, 0.f, 0.f);
    for (int j = beg; j < end; ++j) {
        int s = nbr[j];
        float4 v = xv[(size_t)s * (HIDDEN / 4) + lane];
        acc.x += v.x; acc.y += v.y; acc.z += v.z; acc.w += v.w;
    }
    int deg = end - beg;
    float inv = 1.0f / (float)(deg > 1 ? deg : 1);
    acc.x *= inv; acc.y *= inv; acc.z *= inv; acc.w *= inv;
    ((float4*)mean)[(size_t)wave * (HIDDEN / 4) + lane] = acc;
}

// ---------------- fused dual GEMM via fp32 WMMA ----------------
// out[m,n] = sum_k mean[m,k]*Wl[n,k] + sum_k h[m,k]*Wr[n,k] + b[n]  (opt ReLU)
// One wave computes a 16x128 row strip: 8 N-tiles, 8 v8f accumulators.
// K stepped by 4 (V_WMMA_F32_16X16X4_F32):
//   A 16x4 frag: lanes 0-15 hold K={k,k+1}, lanes 16-31 hold K={k+2,k+3}
//   B 4x16 frag: same K split; n = lane&15  (B[k][n] = W[n][k] -> contiguous)

__global__ __launch_bounds__(256) void sage_gemm(
    const float* __restrict__ mean, const float* __restrict__ h,
    const float* __restrict__ Wl, const float* __restrict__ Wr,
    const float* __restrict__ bias, float* __restrict__ out,
    int M, int relu)
{
    int wave = (blockIdx.x * blockDim.x + threadIdx.x) >> 5;
    int lane = threadIdx.x & 31;
    int mBase = wave * 16;
    if (mBase >= M) return;                 // uniform per wave -> EXEC stays full

    int half = lane >> 4;
    int lr   = lane & 15;

    const float* aRow = mean + (size_t)(mBase + lr) * HIDDEN + 2 * half;
    const float* hRow = h    + (size_t)(mBase + lr) * HIDDEN + 2 * half;
    const float* blRow = Wl + (size_t)lr * HIDDEN + 2 * half;   // + t*16*HIDDEN
    const float* brRow = Wr + (size_t)lr * HIDDEN + 2 * half;

    v8f acc[8];
#pragma unroll
    for (int t = 0; t < 8; ++t) acc[t] = (v8f){};

#pragma unroll 4
    for (int k = 0; k < HIDDEN; k += 4) {
        v2f a  = *(const v2f*)(aRow + k);
        v2f ah = *(const v2f*)(hRow + k);
#pragma unroll
        for (int t = 0; t < 8; ++t) {
            v2f bl = *(const v2f*)(blRow + (size_t)t * 16 * HIDDEN + k);
            v2f br = *(const v2f*)(brRow + (size_t)t * 16 * HIDDEN + k);
            acc[t] = __builtin_amdgcn_wmma_f32_16x16x4_f32(
                false, a, false, bl, (short)0, acc[t], false, false);
            acc[t] = __builtin_amdgcn_wmma_f32_16x16x4_f32(
                false, ah, false, br, (short)0, acc[t], false, false);
        }
    }

#pragma unroll
    for (int t = 0; t < 8; ++t) {
        int nBase = t * 16;
        float bv = bias[nBase + lr];
#pragma unroll
        for (int r = 0; r < 8; ++r) {
            int m = mBase + r + 8 * half;   // C/D layout: VGPR r, lane half
            float v = acc[t][r] + bv;
            if (relu) v = fmaxf(v, 0.f);
            out[(size_t)m * HIDDEN + nBase + lr] = v;
        }
    }
}

// ---------------- host side ----------------

static inline size_t alignup(size_t v) { return (v + 255) & ~(size_t)255; }

extern "C" void kernel_launch(void* const* d_in, const int* in_sizes, int n_in,
                              void* d_out, int out_size, void* d_ws, size_t ws_size,
                              hipStream_t stream)
{
    const float*     x   = (const float*)d_in[0];
    const long long* ei  = (const long long*)d_in[1];   // int64 [2, E]
    const float* Wl1 = (const float*)d_in[2];
    const float* Wr1 = (const float*)d_in[3];
    const float* b1  = (const float*)d_in[4];
    const float* Wl2 = (const float*)d_in[5];
    const float* Wr2 = (const float*)d_in[6];
    const float* b2  = (const float*)d_in[7];
    const float* Wl3 = (const float*)d_in[8];
    const float* Wr3 = (const float*)d_in[9];
    const float* b3  = (const float*)d_in[10];

    const int N = in_sizes[0] / HIDDEN;     // 50000
    const int E = in_sizes[1] / 2;          // 600000

    // workspace carve-up
    char* ws = (char*)d_ws;
    int*   cnt    = (int*)ws;                 ws += alignup((size_t)N * 4);
    int*   cursor = (int*)ws;                 ws += alignup((size_t)N * 4);
    int*   offs   = (int*)ws;                 ws += alignup(((size_t)N + 1) * 4);
    int*   nbr    = (int*)ws;                 ws += alignup((size_t)E * 4);
    float* mean   = (float*)ws;               ws += alignup((size_t)N * HIDDEN * 4);
    float* hA     = (float*)ws;               ws += alignup((size_t)N * HIDDEN * 4);
    float* hB     = (float*)ws;               ws += alignup((size_t)N * HIDDEN * 4);
    (void)ws_size;

    // ---- phase 0: reverse CSR build ----
    hipMemsetAsync(cnt,    0, (size_t)N * 4, stream);
    hipMemsetAsync(cursor, 0, (size_t)N * 4, stream);
    count_kernel<<<(E + 255) / 256, 256, 0, stream>>>(ei, cnt, E);
    scan_kernel<<<1, SCAN_THREADS, 0, stream>>>(cnt, offs, N);
    fill_kernel<<<(E + 255) / 256, 256, 0, stream>>>(ei, offs, cursor, nbr, E);

    // launch geometry
    const int aggBlocks  = (N + 7) / 8;          // 8 waves per 256-thread block
    const int mStrips    = (N + 15) / 16;        // one wave per 16-row strip
    const int gemmBlocks = (mStrips + 7) / 8;

    // ---- layer 1: x -> hA (ReLU) ----
    sage_aggregate<<<aggBlocks, 256, 0, stream>>>(x, offs, nbr, mean, N);
    sage_gemm<<<gemmBlocks, 256, 0, stream>>>(mean, x, Wl1, Wr1, b1, hA, N, 1);

    // ---- layer 2: hA -> hB ----
    sage_aggregate<<<aggBlocks, 256, 0, stream>>>(hA, offs, nbr, mean, N);
    sage_gemm<<<gemmBlocks, 256, 0, stream>>>(mean, hA, Wl2, Wr2, b2, hB, N, 0);

    // ---- layer 3: hB -> d_out ----
    sage_aggregate<<<aggBlocks, 256, 0, stream>>>(hB, offs, nbr, mean, N);
    sage_gemm<<<gemmBlocks, 256, 0, stream>>>(mean, hB, Wl3, Wr3, b3,
                                              (float*)d_out, N, 0);
    (void)out_size; (void)n_in;
}